// KinomeGNN_10720238371005
// MI455X (gfx1250) — compile-verified
//
#include <hip/hip_runtime.h>
#include <math.h>

// Problem constants (match the JAX reference)
#define NN   100000      // total nodes
#define NPG  500         // nodes per graph
#define GG   200         // graphs
#define EE   1600000     // edges
#define HH   32          // hidden dim
#define EMB  16
#define EPSV 1e-5f

typedef __attribute__((ext_vector_type(2))) float v2f;
typedef __attribute__((ext_vector_type(8))) float v8f;

// ---------------------------------------------------------------- zero scratch
__global__ void k_zero(float* __restrict__ p, int n) {
    int i = blockIdx.x * blockDim.x + threadIdx.x;
    if (i < n) p[i] = 0.0f;
}

// ---------------------------------------------------- layer-1 edge scatter-add
// agg1[dst] += x[src]; deg[dst] += 1   (deg reused by layer 2)
__global__ void k_edge1(const float* __restrict__ x,
                        const int* __restrict__ src, const int* __restrict__ dst,
                        float* __restrict__ agg1, float* __restrict__ deg) {
    int e = blockIdx.x * blockDim.x + threadIdx.x;
    if (e >= EE) return;
    __builtin_prefetch(src + e + 2048, 0, 0);   // gfx1250 global_prefetch_b8
    __builtin_prefetch(dst + e + 2048, 0, 0);
    int s = src[e], d = dst[e];
    atomicAdd(&agg1[d], x[s]);
    atomicAdd(&deg[d], 1.0f);
}

// ----------------------------------- layer-1 linear (rank-1) + BN1 stat reduce
// z1[n,c] = (agg1[n]/max(deg,1))*W1l[c] + b1l[c] + x[n]*W1r[c]
// stats[0..31]=sum_c, stats[32..63]=sumsq_c  (atomic per-block partials)
__global__ void k_layer1(const float* __restrict__ x,
                         const float* __restrict__ agg1, const float* __restrict__ deg,
                         const float* __restrict__ W1l, const float* __restrict__ b1l,
                         const float* __restrict__ W1r,
                         float* __restrict__ z1, float* __restrict__ stats) {
    __shared__ float lsum[256], lsq[256];
    int tid  = threadIdx.x;
    int c    = tid & 31;          // channel
    int rloc = tid >> 5;          // wave id in block
    int base = blockIdx.x * 256;  // 256 nodes per block
    float wl = W1l[c], bl = b1l[c], wr = W1r[c];
    float s = 0.0f, sq = 0.0f;
    for (int i = 0; i < 32; ++i) {
        int n = base + rloc * 32 + i;             // wave-uniform node index
        if (n < NN) {
            float dv   = fmaxf(deg[n], 1.0f);
            float mean = agg1[n] / dv;
            float z    = mean * wl + bl + x[n] * wr;
            z1[(size_t)n * HH + c] = z;
            s += z; sq += z * z;
        }
    }
    lsum[tid] = s; lsq[tid] = sq;
    __syncthreads();
    if (tid < 32) {
        float ts = 0.0f, tq = 0.0f;
        for (int r = 0; r < 8; ++r) { ts += lsum[r * 32 + tid]; tq += lsq[r * 32 + tid]; }
        atomicAdd(&stats[tid], ts);
        atomicAdd(&stats[32 + tid], tq);
    }
}

// ------------------------------------------------- BN + ReLU in place (layer 1)
__global__ void k_bnrelu(float* __restrict__ z, const float* __restrict__ stats,
                         const float* __restrict__ gv, const float* __restrict__ bev) {
    int idx = blockIdx.x * blockDim.x + threadIdx.x;
    if (idx >= NN * HH) return;
    int c = idx & 31;
    const float invN = 1.0f / (float)NN;
    float mu  = stats[c] * invN;
    float var = stats[32 + c] * invN - mu * mu;
    float v   = (z[idx] - mu) * rsqrtf(var + EPSV) * gv[c] + bev[c];
    z[idx] = fmaxf(v, 0.0f);
}

// ---------------------------------------------------- layer-2 edge scatter-add
// agg2[dst, 0:32] += h1[src, 0:32]; 8 lanes per edge, float4 each
__global__ void k_edge2(const float* __restrict__ h1,
                        const int* __restrict__ src, const int* __restrict__ dst,
                        float* __restrict__ agg2) {
    int t = blockIdx.x * blockDim.x + threadIdx.x;  // EE*8 threads
    int e = t >> 3, q = t & 7;
    if (e >= EE) return;
    __builtin_prefetch(src + e + 1024, 0, 0);
    int s = src[e], d = dst[e];
    const float4 v = *(const float4*)(h1 + (size_t)s * HH + q * 4);
    float* p = agg2 + (size_t)d * HH + q * 4;
    atomicAdd(p + 0, v.x);
    atomicAdd(p + 1, v.y);
    atomicAdd(p + 2, v.z);
    atomicAdd(p + 3, v.w);
}

// ------------------------------- layer-2 dual GEMM via V_WMMA_F32_16X16X4_F32
// z2 = (agg2/deg) @ W2l + b2l + h1 @ W2r     (z2 written in place over agg2)
// One wave per 16-node tile; K=32 in 8 steps of 4; 2 column tiles; 32 WMMAs.
// BN2 stats (sum/sumsq per channel) reduced out of the D fragments.
__global__ void k_gemm2(float* __restrict__ agg2,      // in: agg2, out: z2
                        const float* __restrict__ h1,
                        const float* __restrict__ deg,
                        const float* __restrict__ W2l, const float* __restrict__ b2l,
                        const float* __restrict__ W2r,
                        float* __restrict__ stats) {
    const int lane  = threadIdx.x & 31;
    const int wid   = threadIdx.x >> 5;
    const int tile  = blockIdx.x * 8 + wid;       // wave-uniform
    if (tile >= NN / 16) return;                  // EXEC stays all-ones for live waves
    const int lcol  = lane & 15;
    const int halfk = lane >> 4;                  // 0: K={k,k+1}; 1: K={k+2,k+3}
    const int row0  = tile * 16;
    const int m     = row0 + lcol;                // A-fragment row for this lane
    const float invd = 1.0f / fmaxf(deg[m], 1.0f);
    const float* Am = agg2 + (size_t)m * HH;      // neighbor-mean source
    const float* Ah = h1   + (size_t)m * HH;      // root source
    const int c0 = lcol, c1 = 16 + lcol;          // output channels (two col tiles)
    const float bz0 = b2l[c0], bz1 = b2l[c1];
    v8f acc0 = {bz0, bz0, bz0, bz0, bz0, bz0, bz0, bz0};
    v8f acc1 = {bz1, bz1, bz1, bz1, bz1, bz1, bz1, bz1};
#pragma unroll
    for (int kb = 0; kb < 8; ++kb) {
        const int k0 = kb * 4 + halfk * 2;
        v2f am;  am.x  = Am[k0] * invd;        am.y  = Am[k0 + 1] * invd;
        v2f ah;  ah.x  = Ah[k0];               ah.y  = Ah[k0 + 1];
        v2f bl0; bl0.x = W2l[k0 * HH + c0];    bl0.y = W2l[(k0 + 1) * HH + c0];
        v2f br0; br0.x = W2r[k0 * HH + c0];    br0.y = W2r[(k0 + 1) * HH + c0];
        v2f bl1; bl1.x = W2l[k0 * HH + c1];    bl1.y = W2l[(k0 + 1) * HH + c1];
        v2f br1; br1.x = W2r[k0 * HH + c1];    br1.y = W2r[(k0 + 1) * HH + c1];
        acc0 = __builtin_amdgcn_wmma_f32_16x16x4_f32(false, am, false, bl0, (short)0, acc0, false, false);
        acc0 = __builtin_amdgcn_wmma_f32_16x16x4_f32(false, ah, false, br0, (short)0, acc0, false, false);
        acc1 = __builtin_amdgcn_wmma_f32_16x16x4_f32(false, am, false, bl1, (short)0, acc1, false, false);
        acc1 = __builtin_amdgcn_wmma_f32_16x16x4_f32(false, ah, false, br1, (short)0, acc1, false, false);
    }
    // Store D and reduce BN2 stats. All agg2 reads above precede these writes;
    // each wave only touches its own 16-row tile, so in-place is safe.
    float s0 = 0.0f, q0 = 0.0f, s1 = 0.0f, q1 = 0.0f;
#pragma unroll
    for (int v = 0; v < 8; ++v) {
        const int row = row0 + v + 8 * halfk;     // C/D layout: lanes 16-31 -> rows 8..15
        const float z0 = acc0[v], z1 = acc1[v];
        agg2[(size_t)row * HH + c0] = z0;
        agg2[(size_t)row * HH + c1] = z1;
        s0 += z0; q0 += z0 * z0;
        s1 += z1; q1 += z1 * z1;
    }
    atomicAdd(&stats[64 + c0], s0);
    atomicAdd(&stats[96 + c0], q0);
    atomicAdd(&stats[64 + c1], s1);
    atomicAdd(&stats[96 + c1], q1);
}

// ----------------------- BN2 + ReLU in place + gate logits (wave per node)
__global__ void k_bnrelu_gate(float* __restrict__ z2, const float* __restrict__ stats,
                              const float* __restrict__ g2, const float* __restrict__ be2,
                              const float* __restrict__ gate_w, const float* __restrict__ gate_b,
                              float* __restrict__ gate) {
    int t = blockIdx.x * blockDim.x + threadIdx.x;
    int n = t >> 5, c = t & 31;
    if (n >= NN) return;
    const float invN = 1.0f / (float)NN;
    float mu  = stats[64 + c] * invN;
    float var = stats[96 + c] * invN - mu * mu;
    float h = fmaxf((z2[(size_t)n * HH + c] - mu) * rsqrtf(var + EPSV) * g2[c] + be2[c], 0.0f);
    z2[(size_t)n * HH + c] = h;
    float p = h * gate_w[c];
#pragma unroll
    for (int o = 16; o > 0; o >>= 1) p += __shfl_xor(p, o, 32);
    if (c == 0) gate[n] = p + gate_b[0];
}

// -------------------------- attention pool: one workgroup per graph (no atomics)
__global__ void k_pool(const float* __restrict__ h2, const float* __restrict__ gate,
                       float* __restrict__ pooled) {
    __shared__ float red[256];
    const int g = blockIdx.x, tid = threadIdx.x;
    const int base = g * NPG;
    // phase 1: max gate
    float m = -3.402823e38f;
    for (int i = tid; i < NPG; i += 256) m = fmaxf(m, gate[base + i]);
    red[tid] = m; __syncthreads();
    for (int s = 128; s > 0; s >>= 1) { if (tid < s) red[tid] = fmaxf(red[tid], red[tid + s]); __syncthreads(); }
    m = red[0]; __syncthreads();
    // phase 2: sum exp
    float se = 0.0f;
    for (int i = tid; i < NPG; i += 256) se += expf(gate[base + i] - m);
    red[tid] = se; __syncthreads();
    for (int s = 128; s > 0; s >>= 1) { if (tid < s) red[tid] += red[tid + s]; __syncthreads(); }
    const float S = red[0]; __syncthreads();
    // phase 3: weighted channel sums
    const int c = tid & 31, r = tid >> 5;
    float acc = 0.0f;
    for (int i = r; i < NPG; i += 8) {
        const int n = base + i;
        acc += h2[(size_t)n * HH + c] * expf(gate[n] - m);
    }
    red[tid] = acc; __syncthreads();
    if (tid < 32) {
        float t = 0.0f;
        for (int r2 = 0; r2 < 8; ++r2) t += red[r2 * 32 + tid];
        pooled[g * HH + tid] = t / S;
    }
}

// ------------------------------------ final MLP + sigmoid: one wave per graph
__global__ void k_mlp(const float* __restrict__ pooled,
                      const int* __restrict__ drug_idx, const int* __restrict__ cell_idx,
                      const float* __restrict__ drug_emb, const float* __restrict__ cell_emb,
                      const float* __restrict__ Wh1, const float* __restrict__ bh1,
                      const float* __restrict__ Wh2, const float* __restrict__ bh2,
                      float* __restrict__ out) {
    const int g = blockIdx.x;
    const int c = threadIdx.x;             // output channel 0..31
    const int di = drug_idx[g], ci = cell_idx[g];
    float h = bh1[c];
#pragma unroll 4
    for (int i = 0; i < HH; ++i)  h += pooled[g * HH + i]      * Wh1[i * HH + c];
#pragma unroll 4
    for (int j = 0; j < EMB; ++j) h += drug_emb[di * EMB + j]  * Wh1[(HH + j) * HH + c];
#pragma unroll 4
    for (int j = 0; j < EMB; ++j) h += cell_emb[ci * EMB + j]  * Wh1[(HH + EMB + j) * HH + c];
    h = fmaxf(h, 0.0f);
    float p = h * Wh2[c];
#pragma unroll
    for (int o = 16; o > 0; o >>= 1) p += __shfl_xor(p, o, 32);
    if (c == 0) out[g] = 1.0f / (1.0f + expf(-(p + bh2[0])));
}

// ---------------------------------------------------------------------- launch
extern "C" void kernel_launch(void* const* d_in, const int* in_sizes, int n_in,
                              void* d_out, int out_size, void* d_ws, size_t ws_size,
                              hipStream_t stream) {
    (void)in_sizes; (void)n_in; (void)out_size; (void)ws_size;
    const float* x        = (const float*)d_in[0];
    const int*   ei       = (const int*)d_in[1];
    const int*   src      = ei;
    const int*   dst      = ei + EE;
    const int*   drug_idx = (const int*)d_in[3];
    const int*   cell_idx = (const int*)d_in[4];
    const float* W1l = (const float*)d_in[5];
    const float* b1l = (const float*)d_in[6];
    const float* W1r = (const float*)d_in[7];
    const float* g1  = (const float*)d_in[8];
    const float* be1 = (const float*)d_in[9];
    const float* W2l = (const float*)d_in[10];
    const float* b2l = (const float*)d_in[11];
    const float* W2r = (const float*)d_in[12];
    const float* g2  = (const float*)d_in[13];
    const float* be2 = (const float*)d_in[14];
    const float* gate_w   = (const float*)d_in[15];
    const float* gate_b   = (const float*)d_in[16];
    const float* drug_emb = (const float*)d_in[17];
    const float* cell_emb = (const float*)d_in[18];
    const float* Wh1 = (const float*)d_in[19];
    const float* bh1 = (const float*)d_in[20];
    const float* Wh2 = (const float*)d_in[21];
    const float* bh2 = (const float*)d_in[22];
    float* out = (float*)d_out;

    // workspace layout (floats)
    float* ws     = (float*)d_ws;
    float* agg1   = ws;                              // N   (zeroed)
    float* deg    = ws + NN;                         // N   (zeroed)
    float* h1buf  = ws + 2 * (size_t)NN;             // N*32: z1 -> h1 in place
    float* agg2   = ws + 2 * (size_t)NN + 32 * (size_t)NN;  // N*32: agg2 -> z2 -> h2 (zeroed)
    float* gate   = ws + 2 * (size_t)NN + 64 * (size_t)NN;  // N
    float* stats  = ws + 3 * (size_t)NN + 64 * (size_t)NN;  // 128 (zeroed)
    float* pooled = stats + 128;                     // G*32

    // zero accumulators every call (harness does not re-poison between replays)
    k_zero<<<(2 * NN + 255) / 256, 256, 0, stream>>>(agg1, 2 * NN);
    k_zero<<<(32 * NN + 255) / 256, 256, 0, stream>>>(agg2, 32 * NN);
    k_zero<<<1, 256, 0, stream>>>(stats, 128);

    k_edge1<<<EE / 256, 256, 0, stream>>>(x, src, dst, agg1, deg);
    k_layer1<<<(NN + 255) / 256, 256, 0, stream>>>(x, agg1, deg, W1l, b1l, W1r, h1buf, stats);
    k_bnrelu<<<(NN * HH) / 256, 256, 0, stream>>>(h1buf, stats, g1, be1);
    k_edge2<<<(EE * 8) / 256, 256, 0, stream>>>(h1buf, src, dst, agg2);
    k_gemm2<<<(NN / 16 + 7) / 8, 256, 0, stream>>>(agg2, h1buf, deg, W2l, b2l, W2r, stats);
    k_bnrelu_gate<<<(NN * HH) / 256, 256, 0, stream>>>(agg2, stats, g2, be2, gate_w, gate_b, gate);
    k_pool<<<GG, 256, 0, stream>>>(agg2, gate, pooled);
    k_mlp<<<GG, 32, 0, stream>>>(pooled, drug_idx, cell_idx, drug_emb, cell_emb,
                                 Wh1, bh1, Wh2, bh2, out);
}